// ChannelwiseVariance_85091892068508
// MI455X (gfx1250) — compile-verified
//
#include <hip/hip_runtime.h>
#include <hip/hip_bf16.h>

// CDNA5 (gfx1250) 3x3 local variance, fp32, zero-padded, divisor 9.
// Vertical box sum expressed as  V = T * X  via V_WMMA_F32_16X16X4_F32
// (T = tridiagonal band-of-ones, generated in registers), horizontal box
// sum done in the C/D layout with intra-half lane shuffles.

typedef __attribute__((ext_vector_type(2))) float v2f;
typedef __attribute__((ext_vector_type(8))) float v8f;

#define IMG_H 512
#define IMG_W 512
#define PLANES (8 * 32)          // B*C image planes
#define TILES_X (IMG_W / 16)     // 32
#define TILES_Y (IMG_H / 16)     // 32
#define TILES_PER_PLANE (TILES_X * TILES_Y) // 1024

__device__ __forceinline__ float band(int m, int k) {
    int d = m - k;
    return (d >= -1 && d <= 1) ? 1.0f : 0.0f;
}

__global__ __launch_bounds__(256)
void ChannelwiseVariance_85091892068508_kernel(const float* __restrict__ x,
                                               float* __restrict__ out) {
    const int wave = (blockIdx.x << 3) + (threadIdx.x >> 5); // 8 waves / block
    const int lane = threadIdx.x & 31;

    const int plane = wave >> 10;            // wave / 1024
    const int tile  = wave & (TILES_PER_PLANE - 1);
    const int tr    = tile >> 5;             // tile row
    const int tc    = tile & (TILES_X - 1);  // tile col
    const int row0  = tr << 4;
    const int col0  = tc << 4;

    const float* __restrict__ px = x   + (long)plane * (IMG_H * IMG_W);
    float*       __restrict__ po = out + (long)plane * (IMG_H * IMG_W);

    const int m  = lane & 15;                // M (A operand) / N (B,C operand)
    const int hi = (lane >= 16) ? 1 : 0;     // which 16-lane half

    // ---- vertical box sums via 4x (K=4) fp32 WMMA, for x and x^2 -------
    // A (16x4): lane m holds rows of the band matrix; VGPR0/1 = K pair,
    //           low half K={4kc,4kc+1}, high half K={4kc+2,4kc+3}.
    // B (4x16): lane n=m holds column n; VGPR0/1 = rows, low half rows
    //           {4kc,4kc+1}, high half {4kc+2,4kc+3}.
    v8f acc  = {};   // boxsum_v(x)
    v8f accq = {};   // boxsum_v(x^2)
#pragma unroll
    for (int kc = 0; kc < 4; ++kc) {
        const int kbase = 4 * kc + 2 * hi;
        v2f a = { band(m, kbase), band(m, kbase + 1) };
        const float b0 = px[(row0 + kbase    ) * IMG_W + col0 + m];
        const float b1 = px[(row0 + kbase + 1) * IMG_W + col0 + m];
        v2f b  = { b0,      b1      };
        v2f bq = { b0 * b0, b1 * b1 };
        // 8 args: (neg_a, A, neg_b, B, c_mod, C, reuse_a, reuse_b)
        acc  = __builtin_amdgcn_wmma_f32_16x16x4_f32(false, a, false, b,
                                                     (short)0, acc,  false, false);
        accq = __builtin_amdgcn_wmma_f32_16x16x4_f32(false, a, false, bq,
                                                     (short)0, accq, false, false);
    }

    // ---- vertical halo: row -1 -> V row 0 (VGPR0 low half),
    //                     row 16 -> V row 15 (VGPR7 high half) ------------
    {
        const int vr = hi ? (row0 + 16) : (row0 - 1);
        const bool ok = (vr >= 0) && (vr < IMG_H);
        const float hv  = ok ? px[vr * IMG_W + col0 + m] : 0.0f;
        const float hvq = hv * hv;
        acc[0]  += hi ? 0.0f : hv;
        accq[0] += hi ? 0.0f : hvq;
        acc[7]  += hi ? hv  : 0.0f;
        accq[7] += hi ? hvq : 0.0f;
    }

    // ---- horizontal halo columns: vertical 3-sum at col-1 (low half,
    //      indexed by row m) and col+16 (high half) -----------------------
    float hs = 0.0f, hsq = 0.0f;
    {
        const int hcol = hi ? (col0 + 16) : (col0 - 1);
        const int hrow = row0 + m;
        if (hcol >= 0 && hcol < IMG_W) {
#pragma unroll
            for (int dr = -1; dr <= 1; ++dr) {
                const int r = hrow + dr;
                if (r >= 0 && r < IMG_H) {
                    const float v = px[r * IMG_W + hcol];
                    hs  += v;
                    hsq += v * v;
                }
            }
        }
    }

    // ---- horizontal pass in C/D layout: columns == lanes ----------------
    const float inv9 = 1.0f / 9.0f;
#pragma unroll
    for (int r = 0; r < 8; ++r) {
        const int rowIdx = r + 8 * hi;       // image row within tile
        const float v  = acc[r];
        const float vq = accq[r];
        // neighbors within the 16-lane half (same image row)
        const float lv  = __shfl_up  (v,  1, 16);
        const float rv  = __shfl_down(v,  1, 16);
        const float lvq = __shfl_up  (vq, 1, 16);
        const float rvq = __shfl_down(vq, 1, 16);
        // halo column sums for this row (lanes 0-15: left, 16-31: right)
        const float lh  = __shfl(hs,  rowIdx);
        const float rh  = __shfl(hs,  16 + rowIdx);
        const float lhq = __shfl(hsq, rowIdx);
        const float rhq = __shfl(hsq, 16 + rowIdx);

        const float sum  = v  + ((m == 0) ? lh  : lv ) + ((m == 15) ? rh  : rv );
        const float sumq = vq + ((m == 0) ? lhq : lvq) + ((m == 15) ? rhq : rvq);

        const float mean = sum * inv9;
        po[(row0 + rowIdx) * IMG_W + col0 + m] = sumq * inv9 - mean * mean;
    }
}

extern "C" void kernel_launch(void* const* d_in, const int* in_sizes, int n_in,
                              void* d_out, int out_size, void* d_ws, size_t ws_size,
                              hipStream_t stream) {
    (void)in_sizes; (void)n_in; (void)out_size; (void)d_ws; (void)ws_size;
    const float* x = (const float*)d_in[0];
    float* out = (float*)d_out;

    const int total_waves = PLANES * TILES_PER_PLANE;  // 262144
    const int block = 256;                             // 8 wave32
    const int grid  = total_waves / (block / 32);      // 32768

    ChannelwiseVariance_85091892068508_kernel<<<grid, block, 0, stream>>>(x, out);
}